// HeteroGATReal_52166672777272
// MI455X (gfx1250) — compile-verified
//
#include <hip/hip_runtime.h>
#include <hip/hip_bf16.h>

#define DIN 256
#define DOUT 128
#define KC 32
#define AS_STRIDE 36     // padded: conflict-free 16-lane float2 A reads, 16B-aligned float4 stores
#define BS2_STRIDE 288   // floats per k2-row (128 float2 pairs + 16 pad pairs); 288%64==32
                         // -> lanes 0-15 (k2) and 16-31 (k2+1) hit disjoint bank halves

typedef __attribute__((ext_vector_type(2))) float v2f;
typedef __attribute__((ext_vector_type(8))) float v8f;

// ---------------------------------------------------------------------------
// GEMM: out[n, 0:128] = X[n, 0:256] @ W + bias, via V_WMMA_F32_16X16X4_F32.
// Block = 256 threads (8 waves). Block tile: 128 rows x 128 cols.
// Wave w owns rows [w*16, w*16+16), all 8 N-tiles (8 x v8f accumulators).
// B staged in pair layout so each B fragment is a single aligned b64 LDS load.
// ---------------------------------------------------------------------------
__global__ __launch_bounds__(256) void gemm_wmma_f32(
    const float* __restrict__ X, const float* __restrict__ W,
    const float* __restrict__ bias, float* __restrict__ out, int nrows)
{
    __shared__ __align__(16) float As[128 * AS_STRIDE];
    __shared__ __align__(16) float Bs2[(KC / 2) * BS2_STRIDE];

    const int tid  = threadIdx.x;
    const int wave = tid >> 5;
    const int lane = tid & 31;
    const int l16  = lane & 15;
    const int kb   = (lane >> 4) << 1;   // 0 for lanes 0-15, 2 for lanes 16-31
    const int khalf = lane >> 4;         // k2 offset: 0 / 1
    const int row0 = blockIdx.x * 128;

    v8f zero = {};
    v8f acc[8];
#pragma unroll
    for (int i = 0; i < 8; ++i) acc[i] = zero;

    for (int kc = 0; kc < DIN; kc += KC) {
        __syncthreads();
        // Stage A chunk: 128 rows x 32 cols. 256 thr * 4 iters * float4.
        {
            const int c = (tid & 7) * 4;
            const int rb = tid >> 3;
#pragma unroll
            for (int it = 0; it < 4; ++it) {
                const int r = rb + it * 32;
                const int gr = row0 + r;
                float4 v = make_float4(0.f, 0.f, 0.f, 0.f);
                if (gr < nrows)
                    v = *(const float4*)(X + (size_t)gr * DIN + kc + c);
                *(float4*)(&As[r * AS_STRIDE + c]) = v;
            }
        }
        // Stage B chunk in pair layout: Bs2[k2][n] = {W[kc+2k2][n], W[kc+2k2+1][n]}.
        // 16 k2-rows x 128 cols of float2. Thread: k2 = t/32 (+8), c = (t%32)*4.
        {
            const int c = (tid & 31) * 4;
            const int k2b = tid >> 5;
#pragma unroll
            for (int it = 0; it < 2; ++it) {
                const int k2 = k2b + it * 8;
                const int gk = kc + 2 * k2;
                float4 v0 = *(const float4*)(W + (size_t)gk * DOUT + c);
                float4 v1 = *(const float4*)(W + (size_t)(gk + 1) * DOUT + c);
                float* dp = &Bs2[k2 * BS2_STRIDE + 2 * c];
                *(float4*)(dp)     = make_float4(v0.x, v1.x, v0.y, v1.y);
                *(float4*)(dp + 4) = make_float4(v0.z, v1.z, v0.w, v1.w);
            }
        }
        __syncthreads();

        const int m = wave * 16 + l16;
#pragma unroll
        for (int kk = 0; kk < KC; kk += 4) {
            // A fragment: lane holds A[m][kk+kb .. kk+kb+1] (contiguous float2).
            v2f a = *(const v2f*)(&As[m * AS_STRIDE + kk + kb]);
            const int k2 = (kk >> 1) + khalf;
            const float* brow = &Bs2[k2 * BS2_STRIDE + 2 * l16];
#pragma unroll
            for (int nt = 0; nt < 8; ++nt) {
                // B fragment: single aligned b64 LDS load, no repacking.
                v2f b = *(const v2f*)(brow + nt * 32);
                acc[nt] = __builtin_amdgcn_wmma_f32_16x16x4_f32(
                    false, a, false, b, (short)0, acc[nt], false, false);
            }
        }
    }

    // Epilogue: C VGPR j -> row (j | j+8 per half-wave), col = nt*16 + l16.
    const int mbase = row0 + wave * 16 + (lane >> 4) * 8;
#pragma unroll
    for (int nt = 0; nt < 8; ++nt) {
        const int col = nt * 16 + l16;
        const float bv = bias[col];
#pragma unroll
        for (int j = 0; j < 8; ++j) {
            const int gr = mbase + j;
            if (gr < nrows)
                out[(size_t)gr * DOUT + col] = acc[nt][j] + bv;
        }
    }
}

// ---------------------------------------------------------------------------
// s[n] = dot(Wh[n,:], a1)  (and optionally a2 -> s2). One wave per node.
// ---------------------------------------------------------------------------
__global__ __launch_bounds__(256) void dot_scores(
    const float* __restrict__ Wh, int n,
    const float* __restrict__ a1, const float* __restrict__ a2,
    float* __restrict__ s1, float* __restrict__ s2)
{
    const int wave = threadIdx.x >> 5;
    const int lane = threadIdx.x & 31;
    const int node = blockIdx.x * 8 + wave;
    if (node >= n) return;

    float4 w = *(const float4*)(Wh + (size_t)node * DOUT + lane * 4);
    float4 x1 = *(const float4*)(a1 + lane * 4);
    float d1 = w.x * x1.x + w.y * x1.y + w.z * x1.z + w.w * x1.w;
    float d2 = 0.f;
    if (s2) {
        float4 x2 = *(const float4*)(a2 + lane * 4);
        d2 = w.x * x2.x + w.y * x2.y + w.z * x2.z + w.w * x2.w;
    }
#pragma unroll
    for (int off = 16; off > 0; off >>= 1) {
        d1 += __shfl_xor(d1, off, 32);
        if (s2) d2 += __shfl_xor(d2, off, 32);
    }
    if (lane == 0) {
        s1[node] = d1;
        if (s2) s2[node] = d2;
    }
}

// ---------------------------------------------------------------------------
// Edge passes. Order-preserving int encoding for float atomic max.
// ---------------------------------------------------------------------------
__device__ __forceinline__ int f32_ord(float f) {
    int b = __float_as_int(f);
    return b >= 0 ? b : (b ^ 0x7fffffff);
}
__device__ __forceinline__ float ord_f32(int o) {
    int b = o >= 0 ? o : (o ^ 0x7fffffff);
    return __int_as_float(b);
}

__global__ void edge_pass1(const int* __restrict__ src, const int* __restrict__ dst,
                           const float* __restrict__ ssrc, const float* __restrict__ sdst,
                           float* __restrict__ ebuf, int* __restrict__ mord, int ne)
{
    const int i = blockIdx.x * 256 + threadIdx.x;
    if (i >= ne) return;
    float e = ssrc[src[i]] + sdst[dst[i]];
    e = e > 0.f ? e : 0.2f * e;
    ebuf[i] = e;
    atomicMax(&mord[dst[i]], f32_ord(e));
}

__global__ void edge_pass2(const int* __restrict__ dst, float* __restrict__ ebuf,
                           const int* __restrict__ mord, float* __restrict__ denom, int ne)
{
    const int i = blockIdx.x * 256 + threadIdx.x;
    if (i >= ne) return;
    const int d = dst[i];
    const int mo = mord[d];
    const float m = (mo == (int)0x80000000) ? 0.f : ord_f32(mo);
    const float ex = expf(ebuf[i] - m);
    ebuf[i] = ex;
    atomicAdd(&denom[d], ex);
}

// One wave per edge: lane covers 4 of 128 channels (b128 load + 4 f32 atomics).
__global__ __launch_bounds__(256) void edge_pass3(
    const int* __restrict__ src, const int* __restrict__ dst,
    const float* __restrict__ exbuf, const float* __restrict__ denom,
    const float* __restrict__ Whmsg, float* __restrict__ out, int ne)
{
    const int wave = threadIdx.x >> 5;
    const int lane = threadIdx.x & 31;
    const int i = blockIdx.x * 8 + wave;
    if (i >= ne) return;
    const int s = src[i];
    const int d = dst[i];
    const float alpha = exbuf[i] / denom[d];
    float4 w = *(const float4*)(Whmsg + (size_t)s * DOUT + lane * 4);
    float* o = out + (size_t)d * DOUT + lane * 4;
    atomicAdd(o + 0, alpha * w.x);
    atomicAdd(o + 1, alpha * w.y);
    atomicAdd(o + 2, alpha * w.z);
    atomicAdd(o + 3, alpha * w.w);
}

__global__ void fill_f32(float* __restrict__ p, float v, int n) {
    const int i = blockIdx.x * 256 + threadIdx.x;
    if (i < n) p[i] = v;
}
__global__ void fill_i32(int* __restrict__ p, int v, int n) {
    const int i = blockIdx.x * 256 + threadIdx.x;
    if (i < n) p[i] = v;
}
__global__ void relu_inplace(float* __restrict__ p, int n) {
    const int i = blockIdx.x * 256 + threadIdx.x;
    if (i < n) p[i] = fmaxf(p[i], 0.f);
}

// ---------------------------------------------------------------------------
extern "C" void kernel_launch(void* const* d_in, const int* in_sizes, int n_in,
                              void* d_out, int out_size, void* d_ws, size_t ws_size,
                              hipStream_t stream)
{
    const float* feat_P = (const float*)d_in[0];
    const float* feat_A = (const float*)d_in[1];
    const int NP = in_sizes[0] / DIN;
    const int NA = in_sizes[1] / DIN;
    const int NE = in_sizes[2];
    const int nmax = NP > NA ? NP : NA;

    // relation order: p2p, p2a, a2p, a2a
    const int* esrc[4] = { (const int*)d_in[2], (const int*)d_in[4],
                           (const int*)d_in[6], (const int*)d_in[8] };
    const int* edst[4] = { (const int*)d_in[3], (const int*)d_in[5],
                           (const int*)d_in[7], (const int*)d_in[9] };
    const float* Wr[4] = { (const float*)d_in[14], (const float*)d_in[16],
                           (const float*)d_in[18], (const float*)d_in[20] };
    const float* Br[4] = { (const float*)d_in[15], (const float*)d_in[17],
                           (const float*)d_in[19], (const float*)d_in[21] };
    const float* ar[4] = { (const float*)d_in[22], (const float*)d_in[23],
                           (const float*)d_in[24], (const float*)d_in[25] };
    const float* fsrc[4] = { feat_P, feat_P, feat_A, feat_A };
    const int    nsrc[4] = { NP, NP, NA, NA };
    const int    ndst[4] = { NP, NA, NP, NA };

    float* outP = (float*)d_out;
    float* outA = outP + (size_t)NP * DOUT;
    float* odst[4] = { outP, outA, outP, outA };

    // Workspace layout (floats): one reusable message matrix + buffers (~61 MB).
    float* ws     = (float*)d_ws;
    float* Whmsg  = ws;                              // nmax*128
    float* sdst   = Whmsg + (size_t)nmax * DOUT;     // 4*nmax (per relation)
    float* ssrc   = sdst + (size_t)4 * nmax;         // nmax
    float* ebuf   = ssrc + nmax;                     // NE
    float* denom  = ebuf + NE;                       // nmax
    int*   mord   = (int*)(denom + nmax);            // nmax

    const dim3 blk(256);
    const int gP  = (NP + 127) / 128;
    const int gA  = (NA + 127) / 128;
    const int gdP = (NP + 7) / 8;
    const int gdA = (NA + 7) / 8;
    const int gE  = (NE + 255) / 256;
    const int gEw = (NE + 7) / 8;

    // 1) Residual base: Wh_P, Wh_A straight into d_out (WMMA GEMM).
    gemm_wmma_f32<<<gP, blk, 0, stream>>>(feat_P, (const float*)d_in[10],
                                          (const float*)d_in[11], outP, NP);
    gemm_wmma_f32<<<gA, blk, 0, stream>>>(feat_A, (const float*)d_in[12],
                                          (const float*)d_in[13], outA, NA);

    // 2) Destination scores (before any scatter mutates d_out).
    //    s_dst: p2p(r0) & a2p(r2) from Wh_P; p2a(r1) & a2a(r3) from Wh_A.
    dot_scores<<<gdP, blk, 0, stream>>>(outP, NP, ar[0] + DOUT, ar[2] + DOUT,
                                        sdst + 0 * (size_t)nmax, sdst + 2 * (size_t)nmax);
    dot_scores<<<gdA, blk, 0, stream>>>(outA, NA, ar[1] + DOUT, ar[3] + DOUT,
                                        sdst + 1 * (size_t)nmax, sdst + 3 * (size_t)nmax);

    // 3) Relations, sequentially, reusing the single message buffer.
    for (int r = 0; r < 4; ++r) {
        const int gS  = (nsrc[r] + 127) / 128;
        const int gDs = (nsrc[r] + 7) / 8;
        const int gF  = (ndst[r] + 255) / 256;

        gemm_wmma_f32<<<gS, blk, 0, stream>>>(fsrc[r], Wr[r], Br[r], Whmsg, nsrc[r]);
        dot_scores<<<gDs, blk, 0, stream>>>(Whmsg, nsrc[r], ar[r], nullptr,
                                            ssrc, nullptr);
        fill_i32<<<gF, blk, 0, stream>>>(mord, (int)0x80000000, ndst[r]);
        fill_f32<<<gF, blk, 0, stream>>>(denom, 0.f, ndst[r]);

        edge_pass1<<<gE, blk, 0, stream>>>(esrc[r], edst[r], ssrc,
                                           sdst + (size_t)r * nmax, ebuf, mord, NE);
        edge_pass2<<<gE, blk, 0, stream>>>(edst[r], ebuf, mord, denom, NE);
        edge_pass3<<<gEw, blk, 0, stream>>>(esrc[r], edst[r], ebuf, denom,
                                            Whmsg, odst[r], NE);
    }

    // 4) h = relu(Wh + recv) in place.
    relu_inplace<<<(out_size + 255) / 256, blk, 0, stream>>>((float*)d_out, out_size);
}